// JBGNN_44942537786116
// MI455X (gfx1250) — compile-verified
//
#include <hip/hip_runtime.h>
#include <hip/hip_bf16.h>

typedef __attribute__((ext_vector_type(2))) float v2f;
typedef __attribute__((ext_vector_type(8))) float v8f;

#define NNODES 50000
#define NEDGES 1600000
#define FIN    256
#define H1F    128
#define H2F    64
#define KC     64
#define EPSV   1e-15f

__device__ __forceinline__ void atomAddF(float* p, float v) {
    unsafeAtomicAdd(p, v);   // lowers to global_atomic_add_f32 (L2 atomic)
}

// ---------------------------------------------------------------------------
// C[M,Nc] = A[M,K] @ B[K,Nc], all row-major fp32.
// One wave (32 lanes) computes one 16x16 tile via V_WMMA_F32_16X16X4_F32.
// Fragment layouts per CDNA5 ISA 7.12.2 (fp32 16x4 A / 4x16 B / 16x16 C).
// ---------------------------------------------------------------------------
__global__ void gemm_wmma_f32(const float* __restrict__ A, const float* __restrict__ B,
                              float* __restrict__ C, int M, int K, int Nc) {
    const int lane = threadIdx.x;
    const int m  = lane & 15;
    const int hi = lane >> 4;            // 0 or 1
    const int tm = blockIdx.x * 16;
    const int tn = blockIdx.y * 16;
    const float* arow = A + (size_t)(tm + m) * K;
    v8f acc = {};
    for (int k = 0; k < K; k += 4) {
        v2f a, b;
        a.x = arow[k + hi * 2];
        a.y = arow[k + hi * 2 + 1];
        b.x = B[(size_t)(k + hi * 2)     * Nc + tn + m];
        b.y = B[(size_t)(k + hi * 2 + 1) * Nc + tn + m];
        acc = __builtin_amdgcn_wmma_f32_16x16x4_f32(false, a, false, b,
                                                    (short)0, acc, false, false);
    }
#pragma unroll
    for (int g = 0; g < 8; ++g)
        C[(size_t)(tm + g + hi * 8) * Nc + tn + m] = acc[g];
}

// ---------------------------------------------------------------------------
// C[64,64] += A^T @ B, A,B: [Ntot,64] row-major fp32. Split-K over node
// chunks (blockIdx.z); partials merged with global float atomics.
// ---------------------------------------------------------------------------
__global__ void atb_wmma_f32(const float* __restrict__ Amat, const float* __restrict__ Bmat,
                             float* __restrict__ C, int Ntot, int chunk) {
    const int lane = threadIdx.x;
    const int m  = lane & 15;
    const int hi = lane >> 4;
    const int tm = blockIdx.x * 16;
    const int tn = blockIdx.y * 16;
    const int k0 = blockIdx.z * chunk;
    const int k1 = min(Ntot, k0 + chunk);
    v8f acc = {};
    for (int k = k0; k < k1; k += 4) {
        v2f a, b;
        a.x = Amat[(size_t)(k + hi * 2)     * KC + tm + m];   // A^T[m][k] = A[k][m]
        a.y = Amat[(size_t)(k + hi * 2 + 1) * KC + tm + m];
        b.x = Bmat[(size_t)(k + hi * 2)     * KC + tn + m];
        b.y = Bmat[(size_t)(k + hi * 2 + 1) * KC + tn + m];
        acc = __builtin_amdgcn_wmma_f32_16x16x4_f32(false, a, false, b,
                                                    (short)0, acc, false, false);
    }
#pragma unroll
    for (int g = 0; g < 8; ++g)
        atomAddF(&C[(size_t)(tm + g + hi * 8) * KC + tn + m], acc[g]);
}

// ---------------------------------------------------------------------------
// Edge scatter-add:  outp[sidx[e], :] += w[e] * h[gidx[e], :]
// One thread per (edge, 4 consecutive features) -> float4 gather, 4 atomics.
// Gathers and atomics stay L2-resident (h and outp fit in 192MB L2).
// ---------------------------------------------------------------------------
template <int F>
__global__ void scatter_add(const float* __restrict__ h, const int* __restrict__ gidx,
                            const int* __restrict__ sidx, const float* __restrict__ w,
                            float* __restrict__ outp) {
    const int Fq  = F / 4;
    int tid = blockIdx.x * blockDim.x + threadIdx.x;
    int e = tid / Fq;
    if (e >= NEDGES) return;
    int q = (tid - e * Fq) * 4;
    int g  = gidx[e];
    int sc = sidx[e];
    float ww = w[e];
    const float4 v = *(const float4*)(h + (size_t)g * F + q);
    float* base = outp + (size_t)sc * F + q;
    atomAddF(base + 0, ww * v.x);
    atomAddF(base + 1, ww * v.y);
    atomAddF(base + 2, ww * v.z);
    atomAddF(base + 3, ww * v.w);
}

// a[i] = relu(a[i] + b[i & (F-1)]), in place. F is a power of two.
__global__ void bias_relu(float* __restrict__ a, const float* __restrict__ b, int total, int Fmask) {
    int i = blockIdx.x * blockDim.x + threadIdx.x;
    if (i >= total) return;
    float v = a[i] + b[i & Fmask];
    a[i] = v > 0.f ? v : 0.f;
}

// Row softmax over 64 logits (+ bias), one wave per node, 2 values per lane.
__global__ void softmax_bias(const float* __restrict__ logits, const float* __restrict__ bias,
                             float* __restrict__ s) {
    int gt = blockIdx.x * blockDim.x + threadIdx.x;
    int node = gt >> 5;
    int lane = gt & 31;
    if (node >= NNODES) return;
    const float* row = logits + (size_t)node * KC;
    float v0 = row[lane * 2]     + bias[lane * 2];
    float v1 = row[lane * 2 + 1] + bias[lane * 2 + 1];
    float mx = fmaxf(v0, v1);
    for (int off = 16; off; off >>= 1) mx = fmaxf(mx, __shfl_xor(mx, off, 32));
    float e0 = __expf(v0 - mx), e1 = __expf(v1 - mx);
    float sm = e0 + e1;
    for (int off = 16; off; off >>= 1) sm += __shfl_xor(sm, off, 32);
    float inv = 1.0f / sm;
    float* orow = s + (size_t)node * KC;
    orow[lane * 2]     = e0 * inv;
    orow[lane * 2 + 1] = e1 * inv;
}

// loss = -sum_i sqrt(ss[i][i] + eps) / sqrt(N*K)
__global__ void loss_kernel(const float* __restrict__ ss, float* __restrict__ loss) {
    __shared__ float red[64];
    int t = threadIdx.x;
    red[t] = sqrtf(ss[t * KC + t] + EPSV);
    __syncthreads();
    for (int off = 32; off; off >>= 1) {
        if (t < off) red[t] += red[t + off];
        __syncthreads();
    }
    if (t == 0) *loss = -red[0] / sqrtf((float)NNODES * (float)KC);
}

// zero diagonal; d = sqrt(rowsum) + eps; adj[i][j] /= d[i]*d[j]  (in place)
__global__ void finalize_adj(float* __restrict__ adj) {
    __shared__ float d[KC];
    int i = threadIdx.x;
    float sum = 0.f;
    for (int j = 0; j < KC; ++j)
        if (j != i) sum += adj[i * KC + j];
    d[i] = sqrtf(sum) + EPSV;
    __syncthreads();
    float di = d[i];
    for (int j = 0; j < KC; ++j) {
        float v = (j == i) ? 0.f : adj[i * KC + j] / (di * d[j]);
        adj[i * KC + j] = v;
    }
}

extern "C" void kernel_launch(void* const* d_in, const int* in_sizes, int n_in,
                              void* d_out, int out_size, void* d_ws, size_t ws_size,
                              hipStream_t stream) {
    const float* x  = (const float*)d_in[0];
    const int*   ei = (const int*)  d_in[1];
    const float* ew = (const float*)d_in[2];
    const float* W1 = (const float*)d_in[3];
    const float* b1 = (const float*)d_in[4];
    const float* W2 = (const float*)d_in[5];
    const float* b2 = (const float*)d_in[6];
    const float* Wc = (const float*)d_in[7];
    const float* bc = (const float*)d_in[8];
    const int* src = ei;             // edge_index[0]
    const int* dst = ei + NEDGES;    // edge_index[1]

    float* outp     = (float*)d_out;
    float* s_out    = outp;                                  // [N,K]
    float* loss_out = outp + (size_t)NNODES * KC;            // scalar
    float* out_pool = loss_out + 1;                          // [K,H2]
    float* adj_out  = out_pool + KC * H2F;                   // [K,K]

    float* bufA = (float*)d_ws;                              // N*128 floats
    float* bufB = bufA + (size_t)NNODES * H1F;               // N*128 floats
    float* ss   = bufB + (size_t)NNODES * H1F;               // 64*64 floats

    // ---- layer 1: h1 = relu(scatter(w * (x@W1)[src] -> dst) + b1) ----
    hipMemsetAsync(bufB, 0, sizeof(float) * (size_t)NNODES * H1F, stream);
    gemm_wmma_f32<<<dim3(NNODES / 16, H1F / 16), 32, 0, stream>>>(x, W1, bufA, NNODES, FIN, H1F);
    {
        int thr = NEDGES * (H1F / 4);
        scatter_add<H1F><<<(thr + 255) / 256, 256, 0, stream>>>(bufA, src, dst, ew, bufB);
    }
    {
        int tot = NNODES * H1F;
        bias_relu<<<(tot + 255) / 256, 256, 0, stream>>>(bufB, b1, tot, H1F - 1);
    }

    // ---- layer 2: h2 = relu(scatter(w * (h1@W2)[src] -> dst) + b2) ----
    gemm_wmma_f32<<<dim3(NNODES / 16, H2F / 16), 32, 0, stream>>>(bufB, W2, bufA, NNODES, H1F, H2F);
    hipMemsetAsync(bufB, 0, sizeof(float) * (size_t)NNODES * H2F, stream);
    {
        int thr = NEDGES * (H2F / 4);
        scatter_add<H2F><<<(thr + 255) / 256, 256, 0, stream>>>(bufA, src, dst, ew, bufB);
    }
    {
        int tot = NNODES * H2F;
        bias_relu<<<(tot + 255) / 256, 256, 0, stream>>>(bufB, b2, tot, H2F - 1);
    }
    // bufB now holds h2 [N,64]

    // ---- s = softmax(h2 @ Wc + bc) ----
    float* logits = bufA + (size_t)NNODES * H2F;
    gemm_wmma_f32<<<dim3(NNODES / 16, KC / 16), 32, 0, stream>>>(bufB, Wc, logits, NNODES, H2F, KC);
    softmax_bias<<<(NNODES * 32 + 255) / 256, 256, 0, stream>>>(logits, bc, s_out);

    // ---- As = scatter(w * s[dst] -> src) ----
    hipMemsetAsync(bufA, 0, sizeof(float) * (size_t)NNODES * H2F, stream);     // As
    hipMemsetAsync(out_pool, 0, sizeof(float) * KC * H2F, stream);
    hipMemsetAsync(adj_out, 0, sizeof(float) * KC * KC, stream);
    hipMemsetAsync(ss, 0, sizeof(float) * KC * KC, stream);
    {
        int thr = NEDGES * (KC / 4);
        scatter_add<KC><<<(thr + 255) / 256, 256, 0, stream>>>(s_out, dst, src, ew, bufA);
    }

    // ---- small reductions: out = s^T h2, ss = s^T s, out_adj = s^T As ----
    const int CH  = 1024;
    const int SPL = (NNODES + CH - 1) / CH;
    atb_wmma_f32<<<dim3(KC / 16, H2F / 16, SPL), 32, 0, stream>>>(s_out, bufB, out_pool, NNODES, CH);
    atb_wmma_f32<<<dim3(KC / 16, KC / 16, SPL), 32, 0, stream>>>(s_out, s_out, ss, NNODES, CH);
    atb_wmma_f32<<<dim3(KC / 16, KC / 16, SPL), 32, 0, stream>>>(s_out, bufA, adj_out, NNODES, CH);

    // ---- loss + adjacency normalization ----
    loss_kernel<<<1, 64, 0, stream>>>(ss, loss_out);
    finalize_adj<<<1, KC, 0, stream>>>(adj_out);
}